// DynamicSepBatchNorm2d_24163486007874
// MI455X (gfx1250) — compile-verified
//
#include <hip/hip_runtime.h>

typedef __attribute__((ext_vector_type(2))) float v2f;
typedef __attribute__((ext_vector_type(4))) float v4f;
typedef __attribute__((ext_vector_type(8))) float v8f;

#define B_    32
#define C_    256
#define HW_   3136      // 56*56
#define HW4_  784       // HW/4
#define HID_  25
#define WAYS_ 3
#define BC_   (B_ * C_) // 8192
#define NTOT_ ((size_t)B_ * C_ * HW_)

// ---------------- Pass 1: per-(b,c) plane sum & sum-of-squares ----------------
__global__ __launch_bounds__(256) void dsbn_reduce(const float* __restrict__ x,
                                                   float* __restrict__ wsum,
                                                   float* __restrict__ wsq) {
    const int bc  = blockIdx.x;                 // 0 .. BC_-1
    const int tid = threadIdx.x;
    const v4f* xp = reinterpret_cast<const v4f*>(x + (size_t)bc * HW_);

    float s = 0.f, q = 0.f;
    for (int i = tid; i < HW4_; i += 256) {
        v4f v = xp[i];                          // regular-temporal: keep x hot in L2
        s += v.x + v.y + v.z + v.w;
        q += v.x * v.x + v.y * v.y + v.z * v.z + v.w * v.w;
    }

    __shared__ float rs[256], rq[256];
    rs[tid] = s; rq[tid] = q;
    __syncthreads();
    for (int off = 128; off > 0; off >>= 1) {
        if (tid < off) { rs[tid] += rs[tid + off]; rq[tid] += rq[tid + off]; }
        __syncthreads();
    }
    if (tid == 0) { wsum[bc] = rs[0]; wsq[bc] = rq[0]; }
}

// ---------------- Pass 2: gating head (WMMA f32) + BN coeff fusion ------------
__global__ __launch_bounds__(256) void dsbn_gate(const float* __restrict__ fc1_w, // [25][256]
                                                 const float* __restrict__ fc2_w, // [3][25]
                                                 const float* __restrict__ gamma, // [3][256]
                                                 const float* __restrict__ beta,  // [3][256]
                                                 const float* __restrict__ wsum,  // [B][C]
                                                 const float* __restrict__ wsq,   // [B][C]
                                                 float* __restrict__ wa,          // [B][C]
                                                 float* __restrict__ wb,          // [B][C]
                                                 float* __restrict__ out) {       // full d_out
    __shared__ float s_pool[B_][C_];     // pooled[b][c]                  (32 KB)
    __shared__ float s_fc1[32][C_];      // fc1 zero-padded to 32 rows    (32 KB)
    __shared__ float s_mean[C_];
    __shared__ float s_istd[C_];
    __shared__ float s_h[B_][32];        // MLP hidden (padded to 32 cols)
    __shared__ float s_logit[B_][WAYS_];
    __shared__ float s_gates[B_][WAYS_];

    const int tid  = threadIdx.x;
    const int lane = tid & 31;
    const int wave = tid >> 5;

    // pooled[b][c] = sum / HW ; stage zero-padded fc1 (rows >= 25 are zero)
    const float inv_hw = 1.0f / (float)HW_;
    for (int i = tid; i < BC_; i += 256) {
        const int r = i >> 8, c = i & 255;
        s_pool[r][c] = wsum[i] * inv_hw;
        s_fc1[r][c]  = (r < HID_) ? fc1_w[r * C_ + c] : 0.0f;
    }
    // per-channel batch mean / inv-std (biased var, as torch BN training)
    {
        const int c = tid;                       // 256 threads == 256 channels
        float s = 0.f, q = 0.f;
        for (int b = 0; b < B_; ++b) {
            s += wsum[b * C_ + c];
            q += wsq [b * C_ + c];
        }
        const float inv_bhw = 1.0f / (float)(B_ * HW_);
        const float m  = s * inv_bhw;
        const float e2 = q * inv_bhw;
        s_mean[c] = m;
        s_istd[c] = rsqrtf(e2 - m * m + 1e-5f);
    }
    __syncthreads();

    // h = pooled @ fc1_w^T  via V_WMMA_F32_16X16X4_F32.
    // 4 waves, each owns one 16x16 (M,N) tile; inner loop: 2x ds_load_b64 + wmma.
    if (wave < 4) {
        const int mt = (wave >> 1) << 4;         // 0 or 16 (rows b)
        const int nt = (wave & 1) << 4;          // 0 or 16 (cols j)
        const int r  = lane & 15;
        const int hi = lane >> 4;                // 0: K{0,1}, 1: K{2,3}
        const int m  = mt + r;                   // A row held by this lane
        const int n  = nt + r;                   // B col held by this lane

        v8f acc = {};
        #pragma unroll 4
        for (int k0 = 0; k0 < C_; k0 += 4) {
            v2f a, b;
            a.x = s_pool[m][k0 + 2 * hi];
            a.y = s_pool[m][k0 + 2 * hi + 1];
            b.x = s_fc1[n][k0 + 2 * hi];
            b.y = s_fc1[n][k0 + 2 * hi + 1];
            acc = __builtin_amdgcn_wmma_f32_16x16x4_f32(
                false, a, false, b, (short)0, acc, false, false);
        }
        // C/D layout: VGPR v -> M = v + 8*hi, N = lane&15 ; apply leaky-relu
        #pragma unroll
        for (int v = 0; v < 8; ++v) {
            float hv = acc[v];
            hv = hv > 0.f ? hv : 0.01f * hv;
            s_h[mt + v + 8 * hi][nt + r] = hv;
        }
    }
    __syncthreads();

    // logits = h @ fc2_w^T, /TEMP
    if (tid < B_ * WAYS_) {
        const int b = tid / WAYS_, w = tid % WAYS_;
        float a = 0.f;
        for (int j = 0; j < HID_; ++j) a += s_h[b][j] * fc2_w[w * HID_ + j];
        s_logit[b][w] = a * (1.0f / 30.0f);
    }
    __syncthreads();

    // softmax over 3 ways; also emit gates to d_out tail
    if (tid < B_) {
        const int b = tid;
        float l0 = s_logit[b][0], l1 = s_logit[b][1], l2 = s_logit[b][2];
        float mx = fmaxf(l0, fmaxf(l1, l2));
        float e0 = expf(l0 - mx), e1 = expf(l1 - mx), e2 = expf(l2 - mx);
        float inv = 1.0f / (e0 + e1 + e2);
        float g0 = e0 * inv, g1 = e1 * inv, g2 = e2 * inv;
        s_gates[b][0] = g0; s_gates[b][1] = g1; s_gates[b][2] = g2;
        float* go = out + NTOT_ + (size_t)b * WAYS_;
        go[0] = g0; go[1] = g1; go[2] = g2;
    }
    __syncthreads();

    // fused per-(b,c) affine: out = x*A + B with
    //   A = istd[c] * (gates@gamma)[b,c];  B = (gates@beta)[b,c] - mean[c]*A
    for (int i = tid; i < BC_; i += 256) {
        const int b = i >> 8, c = i & 255;
        const float g0 = s_gates[b][0], g1 = s_gates[b][1], g2 = s_gates[b][2];
        const float sc = g0 * gamma[c] + g1 * gamma[C_ + c] + g2 * gamma[2 * C_ + c];
        const float bi = g0 * beta [c] + g1 * beta [C_ + c] + g2 * beta [2 * C_ + c];
        const float A  = s_istd[c] * sc;
        wa[i] = A;
        wb[i] = bi - s_mean[c] * A;
    }
}

// ---------------- Pass 3: out = x*A + B (x re-read hits L2; NT stores) --------
__global__ __launch_bounds__(256) void dsbn_apply(const float* __restrict__ x,
                                                  const float* __restrict__ wa,
                                                  const float* __restrict__ wb,
                                                  float* __restrict__ out) {
    const int bc = blockIdx.x;
    const float A  = wa[bc];
    const float Bb = wb[bc];
    const v4f* xp = reinterpret_cast<const v4f*>(x   + (size_t)bc * HW_);
    v4f*       op = reinterpret_cast<v4f*>      (out + (size_t)bc * HW_);
    for (int i = threadIdx.x; i < HW4_; i += 256) {
        v4f v = xp[i];
        v4f r;
        r.x = fmaf(v.x, A, Bb);
        r.y = fmaf(v.y, A, Bb);
        r.z = fmaf(v.z, A, Bb);
        r.w = fmaf(v.w, A, Bb);
        __builtin_nontemporal_store(r, op + i);  // don't evict x from L2
    }
}

extern "C" void kernel_launch(void* const* d_in, const int* in_sizes, int n_in,
                              void* d_out, int out_size, void* d_ws, size_t ws_size,
                              hipStream_t stream) {
    const float* x     = (const float*)d_in[0];
    const float* fc1_w = (const float*)d_in[1];
    const float* fc2_w = (const float*)d_in[2];
    const float* gamma = (const float*)d_in[3];
    const float* beta  = (const float*)d_in[4];
    float* out = (float*)d_out;

    float* wsum = (float*)d_ws;          // [BC_]
    float* wsq  = wsum + BC_;            // [BC_]
    float* wa   = wsq  + BC_;            // [BC_]
    float* wb   = wa   + BC_;            // [BC_]

    dsbn_reduce<<<BC_, 256, 0, stream>>>(x, wsum, wsq);
    dsbn_gate<<<1, 256, 0, stream>>>(fc1_w, fc2_w, gamma, beta, wsum, wsq, wa, wb, out);
    dsbn_apply<<<BC_, 256, 0, stream>>>(x, wa, wb, out);
}